// mid_vgg_direct_residual_wave_47699906789848
// MI455X (gfx1250) — compile-verified
//
#include <hip/hip_runtime.h>
#include <math.h>

typedef __attribute__((ext_vector_type(2))) float v2f;
typedef __attribute__((ext_vector_type(8))) float v8f;

#define WGS 256

__device__ __forceinline__ v8f wmma_f32_16x16x4(v2f a, v2f b, v8f c) {
  // D = A(16x4) * B(4x16) + C(16x16), fp32 WMMA on gfx1250
  return __builtin_amdgcn_wmma_f32_16x16x4_f32(false, a, false, b, (short)0, c, false, false);
}

// ---------------- weight quantization (tanh -> scale by global max -> round) ----------------
// Wsrc: (F, NIN) row-major ; WtQ: (NIN, F)  (i.e. quantized W.T for gather-by-row)
__global__ __launch_bounds__(WGS)
void quantize_kernel(const float* __restrict__ Wsrc, float* __restrict__ WtQ,
                     int F, int NIN) {
  __shared__ float red[WGS];
  const int total = F * NIN;
  float mx = 0.0f;
  for (int i = threadIdx.x; i < total; i += WGS)
    mx = fmaxf(mx, fabsf(tanhf(Wsrc[i])));
  red[threadIdx.x] = mx;
  __syncthreads();
  for (int s = WGS / 2; s > 0; s >>= 1) {
    if (threadIdx.x < s) red[threadIdx.x] = fmaxf(red[threadIdx.x], red[threadIdx.x + s]);
    __syncthreads();
  }
  const float alpha = fmaxf(red[0], 1e-30f);
  for (int i = threadIdx.x; i < total; i += WGS) {
    int f = i / NIN, n = i - f * NIN;
    float t = tanhf(Wsrc[i]) * (1.0f / alpha);
    t = fminf(fmaxf(t, -1.0f), 1.0f) * 127.0f;
    t = rintf(t);                      // round-half-even, matches jnp.round
    WtQ[n * F + f] = t * (alpha / 127.0f);
  }
}

// -------- conv1 (5x5, pad2) + BN + ReLU + maxpool2 + min(.,1) + exp  ->  z1 (8,32,32,32) --------
__global__ __launch_bounds__(WGS)
void conv1_kernel(const float* __restrict__ inp, const float* __restrict__ w,
                  const float* __restrict__ g, const float* __restrict__ bb,
                  const float* __restrict__ bm, const float* __restrict__ bv,
                  float* __restrict__ z1) {
  int t = blockIdx.x * WGS + threadIdx.x;
  if (t >= 8 * 32 * 32 * 32) return;
  int x = t & 31, y = (t >> 5) & 31, c = (t >> 10) & 31, b = t >> 15;
  float scale = g[c] * rsqrtf(bv[c] + 1e-5f);
  float bias  = bb[c] - bm[c] * scale;
  const float* ibase = inp + b * 64 * 64;   // C_in = 1
  const float* wc = w + c * 25;
  float mx = 0.0f;                          // ReLU floor folded into max
  for (int dy = 0; dy < 2; ++dy)
    for (int dx = 0; dx < 2; ++dx) {
      int oy = 2 * y + dy, ox = 2 * x + dx;
      float acc = 0.0f;
      for (int ki = 0; ki < 5; ++ki) {
        int iy = oy - 2 + ki;
        if (iy < 0 || iy >= 64) continue;
        for (int kj = 0; kj < 5; ++kj) {
          int ix = ox - 2 + kj;
          if (ix < 0 || ix >= 64) continue;
          acc += ibase[iy * 64 + ix] * wc[ki * 5 + kj];
        }
      }
      mx = fmaxf(mx, acc * scale + bias);
    }
  mx = fminf(mx, 1.0f);
  z1[t] = expf(mx);
}

// -------- min_pool3: pad-1 (value 1e5), 3x3 window, stride 2 : (.,H,H) -> (.,H/2,H/2) --------
__global__ __launch_bounds__(WGS)
void minpool3_kernel(const float* __restrict__ in, float* __restrict__ out,
                     int H, int OH, int total) {
  int t = blockIdx.x * WGS + threadIdx.x;
  if (t >= total) return;
  int ox = t % OH;
  int rest = t / OH;
  int oy = rest % OH;
  int bc = rest / OH;
  const float* p = in + bc * H * H;
  float mn = 100000.0f;                    // pad value; real data << 1e5
  for (int dy = 0; dy < 3; ++dy) {
    int iy = 2 * oy - 1 + dy;
    if (iy < 0 || iy >= H) continue;
    for (int dx = 0; dx < 3; ++dx) {
      int ix = 2 * ox - 1 + dx;
      if (ix < 0 || ix >= H) continue;
      mn = fminf(mn, p[iy * H + ix]);
    }
  }
  out[t] = mn;
}

__global__ __launch_bounds__(WGS)
void mul_kernel(const float* __restrict__ a, const float* __restrict__ b,
                float* __restrict__ o, int total) {
  int t = blockIdx.x * WGS + threadIdx.x;
  if (t < total) o[t] = a[t] * b[t];
}

// z5c[b, c, y, x] = z5[b, c, y, x] * pm[b, c & 31, y, x]   (channel-concat of pm)
__global__ __launch_bounds__(WGS)
void mulcat_kernel(const float* __restrict__ z5, const float* __restrict__ pm,
                   float* __restrict__ z5c) {
  int t = blockIdx.x * WGS + threadIdx.x;
  if (t >= 8 * 64 * 8 * 8) return;
  int b = t >> 12;           // 64*8*8 = 4096 per batch
  int r = t & 4095;
  int c = r >> 6;
  int yx = r & 63;
  z5c[t] = z5[t] * pm[((b * 32 + (c & 31)) << 6) + yx];
}

// ---------------- sorted-weight gather ----------------
template<bool FC, bool STAGE, int N, int F16, int FOUT>
__device__ __forceinline__ float load_w(const float* __restrict__ Wt,
                                        const float* wl,
                                        const int* sidx, int n, int f) {
  if (FC) {
    // Wt = raw w_fc (FOUT, N); pad f >= FOUT with zeros
    return (f < FOUT) ? Wt[f * N + sidx[n]] : 0.0f;
  } else if (STAGE) {
    // weights staged to LDS by async copy
    return wl[sidx[n] * F16 + f];
  } else {
    // quantized W.T (N_in, F16) from global/L2
    return Wt[sidx[n] * F16 + f];
  }
}

// ================= t_linear_c / t_linear core =================
// One workgroup (8 wave32) per (batch, output position).
// Phases: patch build + async weight stage to LDS -> stable bitonic argsort
//         (overlaps the async copy) -> s_wait_asynccnt -> block sums + carry scan ->
//         WMMA triangular-matmul blocked prefix scan -> spike-time eval -> min reduce.
template<int N, int NP, int F16, int FOUT, bool FC, bool STAGE>
__global__ __launch_bounds__(WGS)
void tlinear_kernel(const float* __restrict__ zin, const float* __restrict__ Wt,
                    float* __restrict__ out,
                    int Cin, int H, int Wsp, int stride, int OH, int OW,
                    float max_spike) {
  constexpr int NB = N / 16;      // 16-row scan blocks
  constexpr int FT = F16 / 16;    // feature tiles
  constexpr int WL = STAGE ? (N * F16) : 4;
  __shared__ float sval[NP];
  __shared__ int   sidx[NP];
  __shared__ float carW[NB * F16];
  __shared__ float carX[NB * F16];
  __shared__ float tmin[NB * F16];
  __shared__ float wlds[WL];

  const int tid = threadIdx.x;
  const int bid = blockIdx.x;
  int b, oy, ox;
  if (FC) { b = bid; oy = 0; ox = 0; }
  else {
    ox = bid % OW;
    oy = (bid / OW) % OH;
    b  = bid / (OW * OH);
  }

  // ---- kick off async weight staging (overlaps with sort below) ----
  if (STAGE) {
    // generic pointer to a __shared__ object: low 32 bits == LDS byte offset
    unsigned lds_base = (unsigned)(uintptr_t)(void*)wlds;
    unsigned long long gbase = (unsigned long long)(uintptr_t)(const void*)Wt;
    constexpr int NV = (N * F16) / 4;          // 16-byte chunks; multiple of WGS
    for (int i = tid; i < NV; i += WGS) {      // uniform trip count: EXEC all-1s
      unsigned voff  = (unsigned)(i * 16);
      unsigned laddr = lds_base + voff;
      asm volatile("global_load_async_to_lds_b128 %0, %1, %2"
                   :: "v"(laddr), "v"(voff), "s"(gbase) : "memory");
    }
  } else {
    // not staged: warm L2/WGP$ while the sort runs (global_prefetch_b8)
    constexpr int WTOT = FC ? (FOUT * N) : (N * F16);
    for (int i = tid * 32; i < WTOT; i += WGS * 32)
      __builtin_prefetch(&Wt[i], 0, 1);
  }

  // ---- build patch (+ sentinel padding to power-of-two) ----
  for (int t = tid; t < NP; t += WGS) {
    float vv;
    if (t < N) {
      if (FC) {
        vv = zin[b * N + t];                       // no <0.1 masking in t_linear
      } else {
        int c  = t / 9;
        int r  = t - c * 9;
        int ki = r / 3;
        int kj = r - ki * 3;
        int iy = oy * stride - 1 + ki;
        int ix = ox * stride - 1 + kj;
        float pv = 0.0f;
        if (iy >= 0 && iy < H && ix >= 0 && ix < Wsp)
          pv = zin[((b * Cin + c) * H + iy) * Wsp + ix];
        vv = (pv < 0.1f) ? max_spike : pv;         // masks pads & sub-threshold
      }
    } else {
      vv = 3.0e38f;                                // sorts past all real data
    }
    sval[t] = vv;
    sidx[t] = t;
  }
  __syncthreads();

  // ---- stable bitonic sort ascending by (value, original index) ----
  for (int k = 2; k <= NP; k <<= 1) {
    for (int j = k >> 1; j > 0; j >>= 1) {
      for (int t = tid; t < NP; t += WGS) {
        int p = t ^ j;
        if (p > t) {
          bool up = ((t & k) == 0);
          float va = sval[t], vb = sval[p];
          int   ia = sidx[t], ib2 = sidx[p];
          bool agtb = (va > vb) || ((va == vb) && (ia > ib2));
          if (agtb == up) {
            sval[t] = vb; sval[p] = va;
            sidx[t] = ib2; sidx[p] = ia;
          }
        }
      }
      __syncthreads();
    }
  }

  // ---- drain async weight copy before first use of wlds ----
  if (STAGE) {
    asm volatile("s_wait_asynccnt 0x0" ::: "memory");
  }
  __syncthreads();

  // ---- per-16-block totals of w and s*w (for inter-block carries) ----
  for (int pq = tid; pq < NB * F16; pq += WGS) {
    int nb = pq / F16;
    int f  = pq - nb * F16;
    float sw = 0.0f, sx = 0.0f;
    for (int jj = 0; jj < 16; ++jj) {
      int n = nb * 16 + jj;
      float w = load_w<FC, STAGE, N, F16, FOUT>(Wt, wlds, sidx, n, f);
      sw += w;
      sx += sval[n] * w;
    }
    carW[pq] = sw;
    carX[pq] = sx;
  }
  __syncthreads();
  // exclusive scan over blocks, per feature (in place)
  if (tid < F16) {
    float rw = 0.0f, rx = 0.0f;
    for (int nb = 0; nb < NB; ++nb) {
      int pq = nb * F16 + tid;
      float tw = carW[pq], tx = carX[pq];
      carW[pq] = rw; carX[pq] = rx;
      rw += tw; rx += tx;
    }
  }
  __syncthreads();

  // ---- WMMA blocked inclusive scan: D = L16 * X  via 4 chained 16x16x4 f32 WMMAs ----
  const int lane = tid & 31;
  const int wave = tid >> 5;
  const int hi = lane >> 4;     // K-half select for A/B fragments
  const int fl = lane & 15;     // = A-row (M) and B-column (feature) for this lane
  for (int tile = wave; tile < NB * FT; tile += WGS / 32) {   // uniform per wave: EXEC all-1s
    int nb = tile / FT;
    int ft = tile - nb * FT;
    int f  = ft * 16 + fl;
    v8f dw = {0.f, 0.f, 0.f, 0.f, 0.f, 0.f, 0.f, 0.f};
    v8f dx = {0.f, 0.f, 0.f, 0.f, 0.f, 0.f, 0.f, 0.f};
#pragma unroll
    for (int c = 0; c < 4; ++c) {
      int k0 = 4 * c + 2 * hi;            // A: VGPR0 holds K=k0, VGPR1 holds K=k0+1
      int n0 = nb * 16 + k0;
      v2f a, bw, bx;
      a.x = (k0     <= fl) ? 1.0f : 0.0f; // lower-triangular ones (inclusive scan)
      a.y = (k0 + 1 <= fl) ? 1.0f : 0.0f;
      float w0 = load_w<FC, STAGE, N, F16, FOUT>(Wt, wlds, sidx, n0,     f);
      float w1 = load_w<FC, STAGE, N, F16, FOUT>(Wt, wlds, sidx, n0 + 1, f);
      bw.x = w0;             bw.y = w1;
      bx.x = sval[n0] * w0;  bx.y = sval[n0 + 1] * w1;
      dw = wmma_f32_16x16x4(a, bw, dw);
      dx = wmma_f32_16x16x4(a, bx, dx);
    }
    float cw = carW[nb * F16 + f];
    float cx = carX[nb * F16 + f];
    float amin = 3.0e38f;
#pragma unroll
    for (int r = 0; r < 8; ++r) {
      int mrow = r + 8 * hi;              // C/D layout: VGPR r = row r (lanes 0-15) / r+8
      int n = nb * 16 + mrow;
      float wsum  = dw[r] + cw;
      float iwsum = dx[r] + cx;
      float sa  = sval[n];
      float nxt = (n == N - 1) ? 1.0f : sval[n + 1];
      float den = fminf(fmaxf(wsum - 1.0f, 1e-10f), 1e10f);
      float o = iwsum / den;
      if (wsum < 1.0f) o = max_spike;
      if (o < sa)      o = max_spike;
      bool late = FC ? (o >= nxt) : (o > nxt);   // t_linear uses >=, t_linear_c uses >
      if (late)        o = max_spike;
      amin = fminf(amin, o);
    }
    amin = fminf(amin, __shfl_xor(amin, 16, 32));  // combine rows 0-7 with 8-15 (same f)
    if (hi == 0) tmin[nb * F16 + f] = amin;
  }
  __syncthreads();

  // ---- final min over scan blocks, write output ----
  if (tid < FOUT) {
    float mn = 3.0e38f;
    for (int nb = 0; nb < NB; ++nb) mn = fminf(mn, tmin[nb * F16 + tid]);
    if (FC) out[b * FOUT + tid] = mn;
    else    out[((b * FOUT + tid) * OH + oy) * OW + ox] = mn;
  }
}

extern "C" void kernel_launch(void* const* d_in, const int* in_sizes, int n_in,
                              void* d_out, int out_size, void* d_ws, size_t ws_size,
                              hipStream_t stream) {
  const float* inp = (const float*)d_in[0];
  const float* c1w = (const float*)d_in[1];
  const float* bng = (const float*)d_in[2];
  const float* bnb = (const float*)d_in[3];
  const float* bnm = (const float*)d_in[4];
  const float* bnv = (const float*)d_in[5];
  const float* w2  = (const float*)d_in[6];
  const float* w3  = (const float*)d_in[7];
  const float* w4  = (const float*)d_in[8];
  const float* w5  = (const float*)d_in[9];
  const float* wfc = (const float*)d_in[10];
  float* out = (float*)d_out;
  float* ws  = (float*)d_ws;

  // workspace layout (floats), ~2.85 MB total
  float* z1  = ws;             // 8*32*32*32 = 262144
  float* m1  = z1  + 262144;   // 8*32*16*16 =  65536
  float* z2  = m1  + 65536;    //               65536
  float* z3  = z2  + 65536;    //               65536
  float* z3c = z3  + 65536;    //               65536
  float* pm  = z3c + 65536;    // 8*32*8*8   =  16384
  float* z4  = pm  + 16384;    // 8*64*8*8   =  32768
  float* z5  = z4  + 32768;    //               32768
  float* z5c = z5  + 32768;    //               32768
  float* q2  = z5c + 32768;    // 288*32     =   9216
  float* q3  = q2  + 9216;     //                9216
  float* q4  = q3  + 9216;     // 288*64     =  18432
  float* q5  = q4  + 18432;    // 576*64     =  36864

  const float MZ = 2.7182818284590452f;   // float(np.e)

  quantize_kernel<<<1, WGS, 0, stream>>>(w2, q2, 32, 288);
  quantize_kernel<<<1, WGS, 0, stream>>>(w3, q3, 32, 288);
  quantize_kernel<<<1, WGS, 0, stream>>>(w4, q4, 64, 288);
  quantize_kernel<<<1, WGS, 0, stream>>>(w5, q5, 64, 576);

  conv1_kernel<<<(8 * 32 * 32 * 32) / WGS, WGS, 0, stream>>>(inp, c1w, bng, bnb, bnm, bnv, z1);
  minpool3_kernel<<<65536 / WGS, WGS, 0, stream>>>(z1, m1, 32, 16, 65536);

  // z2 = t_conv(z1, w2, k3 p1 s2, MAX_Z)   -- weights async-staged to LDS (36 KB)
  tlinear_kernel<288, 512, 32, 32, false, true><<<8 * 16 * 16, WGS, 0, stream>>>(
      z1, q2, z2, 32, 32, 32, 2, 16, 16, MZ);
  // z3 = t_conv(z2, w3, k3 p1 s1, MAX_Z)
  tlinear_kernel<288, 512, 32, 32, false, true><<<8 * 16 * 16, WGS, 0, stream>>>(
      z2, q3, z3, 32, 16, 16, 1, 16, 16, MZ);

  mul_kernel<<<65536 / WGS, WGS, 0, stream>>>(z3, m1, z3c, 65536);
  minpool3_kernel<<<16384 / WGS, WGS, 0, stream>>>(z3c, pm, 16, 8, 16384);

  // z4 = t_conv(z3c, w4, k3 p1 s2, 2*MAX_Z) -- weights async-staged to LDS (72 KB)
  tlinear_kernel<288, 512, 64, 64, false, true><<<8 * 8 * 8, WGS, 0, stream>>>(
      z3c, q4, z4, 32, 16, 16, 2, 8, 8, 2.0f * MZ);
  // z5 = t_conv(z4, w5, k3 p1 s1, 2*MAX_Z)  -- weights too large to stage; prefetch to L2
  tlinear_kernel<576, 1024, 64, 64, false, false><<<8 * 8 * 8, WGS, 0, stream>>>(
      z4, q5, z5, 64, 8, 8, 1, 8, 8, 2.0f * MZ);

  mulcat_kernel<<<32768 / WGS, WGS, 0, stream>>>(z5, pm, z5c);

  // out = t_linear(z5c, w_fc, 4*MAX_Z)  -> (8, 10)
  tlinear_kernel<4096, 4096, 16, 10, true, false><<<8, WGS, 0, stream>>>(
      z5c, wfc, out, 0, 0, 0, 0, 0, 0, 4.0f * MZ);
}